// Encoder_MPGINv2_58007828300368
// MI455X (gfx1250) — compile-verified
//
#include <hip/hip_runtime.h>
#include <hip/hip_bf16.h>

typedef __attribute__((ext_vector_type(2))) float v2f;
typedef __attribute__((ext_vector_type(8))) float v8f;

// ---------------------------------------------------------------------------
// GEMM: D[M,128] = opt_relu( (A1 + A2?) [M,K] @ W[K,128] + bias )
// One wave computes a 16-row strip across all 8 column tiles using
// V_WMMA_F32_16X16X4_F32. Weights staged through LDS in K=32 chunks
// (vectorized b128 copy + global_prefetch of the next chunk).
// fp32 WMMA layout:
//   A 16x4: lanes 0-15 hold K=k,k+1 (vgpr0,1); lanes 16-31 hold K=k+2,k+3
//   B 4x16: row (fixed K) striped across lanes; same K<->(vgpr,half) map as A
//   C/D 16x16: vgpr r, lane half h, lane lr -> element [r + 8h][lr]
// ---------------------------------------------------------------------------
__global__ __launch_bounds__(256)
void gemm128_wmma(const float* __restrict__ A1, long ldA1,
                  const float* __restrict__ A2, long ldA2,
                  const float* __restrict__ W,      // K x 128 row-major
                  const float* __restrict__ bias,   // 128
                  float* __restrict__ D, long ldD,
                  int M, int K, int relu_out) {
  __shared__ __align__(16) float sW[32 * 128];      // 16 KB chunk of W
  const int lane = threadIdx.x & 31;
  const int wib  = threadIdx.x >> 5;                // wave in block, 0..7
  const int half = lane >> 4;
  const int lr   = lane & 15;
  const long mtile = (long)blockIdx.x * 8 + wib;
  const long row   = mtile * 16 + lr;
  const long rowc  = (row < (long)M) ? row : (long)M - 1;   // clamp tail

  v8f acc[8];
#pragma unroll
  for (int nt = 0; nt < 8; ++nt) acc[nt] = (v8f)(0.0f);

  const float* aRow1 = A1 + rowc * ldA1;
  const float* aRow2 = A2 ? (A2 + rowc * ldA2) : nullptr;

  for (int kc = 0; kc < K; kc += 32) {
    // ---- stage W[kc..kc+32) x 128 into LDS, float4 per thread ----
    {
      const float4* wsrc = (const float4*)(W + (long)kc * 128);
      float4*       sdst = (float4*)sW;
#pragma unroll
      for (int p = 0; p < 4; ++p)
        sdst[p * 256 + threadIdx.x] = wsrc[p * 256 + threadIdx.x];
      if (kc + 32 < K)   // prefetch next chunk (global_prefetch_b8)
        __builtin_prefetch(W + (long)(kc + 32) * 128 + threadIdx.x * 16, 0, 1);
    }
    __syncthreads();
#pragma unroll
    for (int k4 = 0; k4 < 32; k4 += 4) {
      const int  ks = k4 + 2 * half;                // K offset inside chunk
      const long kk = (long)kc + ks;
      v2f a = *(const v2f*)(aRow1 + kk);            // 8B aligned (kk even)
      if (aRow2) {
        v2f a2 = *(const v2f*)(aRow2 + kk);
        a.x += a2.x; a.y += a2.y;
      }
#pragma unroll
      for (int nt = 0; nt < 8; ++nt) {
        v2f b;
        b.x = sW[(ks + 0) * 128 + nt * 16 + lr];
        b.y = sW[(ks + 1) * 128 + nt * 16 + lr];
        acc[nt] = __builtin_amdgcn_wmma_f32_16x16x4_f32(
            false, a, false, b, (short)0, acc[nt], false, false);
      }
    }
    __syncthreads();
  }

#pragma unroll
  for (int nt = 0; nt < 8; ++nt) {
#pragma unroll
    for (int r = 0; r < 8; ++r) {
      const long orow = mtile * 16 + r + 8 * half;
      if (orow < (long)M) {
        const int col = nt * 16 + lr;
        float v = acc[nt][r] + bias[col];
        if (relu_out) v = fmaxf(v, 0.0f);
        D[orow * ldD + col] = v;
      }
    }
  }
}

// ---------------------------------------------------------------------------
// GINE edge message: agg[dst] += relu(x[src] + edge_attr). Wave per edge.
// ---------------------------------------------------------------------------
__global__ __launch_bounds__(256)
void edge_msg(const float* __restrict__ x, long ldx,
              const float* __restrict__ ea,
              const int* __restrict__ src, const int* __restrict__ dst,
              float* __restrict__ agg, int nE) {
  const int gw   = (int)(((long)blockIdx.x * 256 + threadIdx.x) >> 5);
  const int lane = threadIdx.x & 31;
  if (gw >= nE) return;
  const long s = src[gw], d = dst[gw];
  const float4 xv = *(const float4*)(x + s * ldx + lane * 4);
  const float4 ev = *(const float4*)(ea + (long)gw * 128 + lane * 4);
  float* p = agg + d * 128 + lane * 4;
  atomicAdd(p + 0, fmaxf(xv.x + ev.x, 0.0f));
  atomicAdd(p + 1, fmaxf(xv.y + ev.y, 0.0f));
  atomicAdd(p + 2, fmaxf(xv.z + ev.z, 0.0f));
  atomicAdd(p + 3, fmaxf(xv.w + ev.w, 0.0f));
}

// GIN motif edge scatter: magg[dst] += h[src]. Wave per edge.
__global__ __launch_bounds__(256)
void motif_scatter(const float* __restrict__ h, long ldh,
                   const int* __restrict__ src, const int* __restrict__ dst,
                   float* __restrict__ magg, int nE) {
  const int gw   = (int)(((long)blockIdx.x * 256 + threadIdx.x) >> 5);
  const int lane = threadIdx.x & 31;
  if (gw >= nE) return;
  const long s = src[gw], d = dst[gw];
  const float4 hv = *(const float4*)(h + s * ldh + lane * 4);
  float* p = magg + d * 128 + lane * 4;
  atomicAdd(p + 0, hv.x);
  atomicAdd(p + 1, hv.y);
  atomicAdd(p + 2, hv.z);
  atomicAdd(p + 3, hv.w);
}

// node -> motif scatter: xmh[node2motif[i] + partial[batch[i]]] += xmn[i]
__global__ __launch_bounds__(256)
void node_motif_scatter(const float* __restrict__ xmn,
                        const int* __restrict__ n2m,
                        const int* __restrict__ batch,
                        const int* __restrict__ partial,
                        float* __restrict__ xmh, int nN) {
  const int gw   = (int)(((long)blockIdx.x * 256 + threadIdx.x) >> 5);
  const int lane = threadIdx.x & 31;
  if (gw >= nN) return;
  const long mid = (long)n2m[gw] + partial[batch[gw]];
  const float4 v = *(const float4*)(xmn + (long)gw * 128 + lane * 4);
  float* p = xmh + mid * 128 + lane * 4;
  atomicAdd(p + 0, v.x);
  atomicAdd(p + 1, v.y);
  atomicAdd(p + 2, v.z);
  atomicAdd(p + 3, v.w);
}

// exclusive prefix of num_motifs (B is tiny: 64)
__global__ void scan_partial(const int* __restrict__ nm, int* __restrict__ partial, int Bn) {
  const int b = threadIdx.x;
  if (b < Bn) {
    int s = 0;
    for (int i = 0; i < b; ++i) s += nm[i];
    partial[b] = s;
  }
}

// motif_batch[m] = b with partial[b] <= m < partial[b]+nm[b]
__global__ __launch_bounds__(256)
void motif_batch_kernel(const int* __restrict__ partial, const int* __restrict__ nm,
                        int* __restrict__ mb, int MT, int Bn) {
  const int m = blockIdx.x * 256 + threadIdx.x;
  if (m >= MT) return;
  int b = Bn - 1;
  for (int i = 0; i < Bn; ++i) {
    if (m >= partial[i] && m < partial[i] + nm[i]) { b = i; break; }
  }
  mb[m] = b;
}

// xmh[m][c] += emb[motifid[m]][c]
__global__ __launch_bounds__(256)
void add_emb(float* __restrict__ xmh, const int* __restrict__ motifid,
             const float* __restrict__ emb, int MT) {
  const long idx = (long)blockIdx.x * 256 + threadIdx.x;
  if (idx >= (long)MT * 128) return;
  const long m = idx >> 7;
  const int  c = (int)(idx & 127);
  xmh[idx] += emb[(long)motifid[m] * 128 + c];
}

// dst[seg[r]][c] += src[r][c]  (contiguous rows x cols)
__global__ __launch_bounds__(256)
void pool_rows(const float* __restrict__ src, const int* __restrict__ seg,
               float* __restrict__ dst, long rows, int cols) {
  const long idx = (long)blockIdx.x * 256 + threadIdx.x;
  if (idx >= rows * (long)cols) return;
  const long r = idx / cols;
  const int  c = (int)(idx % cols);
  atomicAdd(dst + (long)seg[r] * cols + c, src[idx]);
}

// ---------------------------------------------------------------------------
extern "C" void kernel_launch(void* const* d_in, const int* in_sizes, int n_in,
                              void* d_out, int out_size, void* d_ws, size_t ws_size,
                              hipStream_t stream) {
  const float* x_in   = (const float*)d_in[0];
  const float* eattr  = (const float*)d_in[1];
  const int*   eidx   = (const int*)d_in[2];
  const int*   batch  = (const int*)d_in[3];
  const int*   n2m    = (const int*)d_in[4];
  const int*   nummot = (const int*)d_in[5];
  const int*   meidx  = (const int*)d_in[6];
  const int*   motifid= (const int*)d_in[7];
  const float* gc_W1  = (const float*)d_in[8];
  const float* gc_b1  = (const float*)d_in[9];
  const float* gc_W2  = (const float*)d_in[10];
  const float* gc_b2  = (const float*)d_in[11];
  const float* mc_W1  = (const float*)d_in[12];
  const float* mc_b1  = (const float*)d_in[13];
  const float* mc_W2  = (const float*)d_in[14];
  const float* mc_b2  = (const float*)d_in[15];
  const float* lin_W  = (const float*)d_in[16];
  const float* lin_b  = (const float*)d_in[17];
  const float* emb    = (const float*)d_in[18];

  const int N   = in_sizes[0] / 128;          // nodes
  const int E   = in_sizes[2] / 2;            // edges
  const int Bn  = in_sizes[5];                // graphs
  const int MT  = in_sizes[7];                // total motifs
  const int EM  = in_sizes[6] / 2;            // motif edges
  const int NGC = in_sizes[8]  / (128 * 128); // 3
  const int NMC = in_sizes[12] / (128 * 128); // 2

  float* dout = (float*)d_out;

  // workspace layout
  float* ws = (float*)d_ws;
  size_t o = 0;
  float* xcat  = ws + o; o += (size_t)N * 384;   // concat of 3 GINE outputs
  float* agg   = ws + o; o += (size_t)N * 128;
  float* htmp  = ws + o; o += (size_t)N * 128;   // GEMM1 out / xm_nodes
  float* xmh   = ws + o; o += (size_t)MT * 128;
  float* magg  = ws + o; o += (size_t)MT * 128;
  float* mtmp  = ws + o; o += (size_t)MT * 128;
  float* hmcat = ws + o; o += (size_t)MT * 256;  // concat of 2 GIN outputs
  int* partial = (int*)(ws + o); o += Bn;
  int* mb      = (int*)(ws + o); o += MT;

  hipMemsetAsync(d_out, 0, (size_t)out_size * sizeof(float), stream);

  const int* esrc = eidx;
  const int* edst = eidx + E;
  const int gemmGridN = (int)(((((long)N + 15) / 16) + 7) / 8);
  const int gemmGridM = (int)(((((long)MT + 15) / 16) + 7) / 8);

  // ---- GINE stack ----
  const float* xprev = x_in; long ldx = 128;
  for (int i = 0; i < NGC; ++i) {
    hipMemsetAsync(agg, 0, (size_t)N * 128 * sizeof(float), stream);
    edge_msg<<<(E + 7) / 8, 256, 0, stream>>>(xprev, ldx, eattr, esrc, edst, agg, E);
    gemm128_wmma<<<gemmGridN, 256, 0, stream>>>(
        xprev, ldx, agg, 128, gc_W1 + (size_t)i * 128 * 128, gc_b1 + (size_t)i * 128,
        htmp, 128, N, 128, 1);
    gemm128_wmma<<<gemmGridN, 256, 0, stream>>>(
        htmp, 128, nullptr, 0, gc_W2 + (size_t)i * 128 * 128, gc_b2 + (size_t)i * 128,
        xcat + (size_t)i * 128, 384, N, 128, 1);
    xprev = xcat + (size_t)i * 128; ldx = 384;
  }

  // xg = segment_sum(xcat, batch) -> dout[Bn*256 ..]
  {
    long tot = (long)N * 384;
    pool_rows<<<(int)((tot + 255) / 256), 256, 0, stream>>>(
        xcat, batch, dout + (size_t)Bn * 256, N, 384);
  }

  // xm_nodes = relu(xcat @ lin_W + lin_b) -> htmp
  gemm128_wmma<<<gemmGridN, 256, 0, stream>>>(
      xcat, 384, nullptr, 0, lin_W, lin_b, htmp, 128, N, 384, 1);

  // motif offsets / batch ids
  scan_partial<<<1, 128, 0, stream>>>(nummot, partial, Bn);
  motif_batch_kernel<<<(MT + 255) / 256, 256, 0, stream>>>(partial, nummot, mb, MT, Bn);

  // xm_h = segment_sum(xm_nodes, n2m+partial[batch]) + emb[motifid]
  hipMemsetAsync(xmh, 0, (size_t)MT * 128 * sizeof(float), stream);
  node_motif_scatter<<<(N + 7) / 8, 256, 0, stream>>>(htmp, n2m, batch, partial, xmh, N);
  {
    long tot = (long)MT * 128;
    add_emb<<<(int)((tot + 255) / 256), 256, 0, stream>>>(xmh, motifid, emb, MT);
  }

  // ---- motif GIN stack ----
  const int* msrc = meidx;
  const int* mdst = meidx + EM;
  const float* hprev = xmh; long ldh = 128;
  for (int i = 0; i < NMC; ++i) {
    hipMemsetAsync(magg, 0, (size_t)MT * 128 * sizeof(float), stream);
    motif_scatter<<<(EM + 7) / 8, 256, 0, stream>>>(hprev, ldh, msrc, mdst, magg, EM);
    gemm128_wmma<<<gemmGridM, 256, 0, stream>>>(
        hprev, ldh, magg, 128, mc_W1 + (size_t)i * 128 * 128, mc_b1 + (size_t)i * 128,
        mtmp, 128, MT, 128, 1);
    gemm128_wmma<<<gemmGridM, 256, 0, stream>>>(
        mtmp, 128, nullptr, 0, mc_W2 + (size_t)i * 128 * 128, mc_b2 + (size_t)i * 128,
        hmcat + (size_t)i * 128, 256, MT, 128, 1);
    hprev = hmcat + (size_t)i * 128; ldh = 256;
  }

  // xm = segment_sum(hmcat, motif_batch) -> dout[0 ..]
  {
    long tot = (long)MT * 256;
    pool_rows<<<(int)((tot + 255) / 256), 256, 0, stream>>>(hmcat, mb, dout, MT, 256);
  }
}